// ClusterNet_7000796692866
// MI455X (gfx1250) — compile-verified
//
#include <hip/hip_runtime.h>

// ClusterNet soft-assignment (Q) + target distribution (P) for MI455X (gfx1250).
//
// sim^2[i,k] = ||z_i||^2 + ||c_k||^2 - 2 * dot(z_i, c_k)
// Q = rownorm(1/(1+sim)); P = rownorm(Q^2 / colsum(Q))
//
// HBM-bound (~14 MB traffic -> ~0.6us floor at 23.3 TB/s): keep full f32
// precision, V_WMMA_F32_16X16X4_F32 for the 8192x64x256 GEMM.
// Wave32: one wave owns a 16x64 output strip (four 16x16 f32 accumulators).
// Centroid norms are precomputed (64 KB re-read, negligible) to keep the
// hot K-loop free of extra VALU chains and register pressure.

typedef __attribute__((ext_vector_type(2))) float v2f;
typedef __attribute__((ext_vector_type(8))) float v8f;

namespace {
constexpr int KD    = 256;   // feature dim (K)
constexpr int NC    = 64;    // clusters (N)
constexpr int NR    = 8192;  // rows (M)
constexpr int WAVES = 8;     // waves per block (256 threads)
constexpr int QBLK  = NR / (WAVES * 16);  // 64 blocks for the Q kernel
}

// ---------------------------------------------------------------------------
// Kernel 0: cnorm[k] = ||centroid_k||^2. One wave per centroid row.
// ---------------------------------------------------------------------------
__global__ void __launch_bounds__(256)
clusternet_cnorm_kernel(const float* __restrict__ cent,
                        float* __restrict__ cnorm) {
    const int lane = threadIdx.x & 31;
    const int row  = blockIdx.x * WAVES + (threadIdx.x >> 5);
    const float* cr = cent + (size_t)row * KD;

    float s = 0.f;
#pragma unroll
    for (int h = 0; h < KD; h += 32) {
        float v = cr[h + lane];
        s += v * v;
    }
    s += __shfl_xor(s, 1, 32);
    s += __shfl_xor(s, 2, 32);
    s += __shfl_xor(s, 4, 32);
    s += __shfl_xor(s, 8, 32);
    s += __shfl_xor(s, 16, 32);
    if (lane == 0) cnorm[row] = s;
}

// ---------------------------------------------------------------------------
// Kernel 1: Q (row-normalized) + per-block column partial sums (deterministic)
// ---------------------------------------------------------------------------
__global__ void __launch_bounds__(256)
clusternet_q_kernel(const float* __restrict__ z,
                    const float* __restrict__ cent,
                    const float* __restrict__ cnorm,
                    float* __restrict__ Qout,
                    float* __restrict__ blockpart) {
    __shared__ float csum[WAVES][NC];

    const int lane = threadIdx.x & 31;
    const int wave = threadIdx.x >> 5;
    const int half = lane >> 4;   // which 16-lane half of the wave
    const int n16  = lane & 15;   // column index within a 16-wide tile
    const int rowBase = (blockIdx.x * WAVES + wave) * 16;

    // A fragment source: row (rowBase+n16), K split 2+2 across half-waves.
    const float* zrow = z + (size_t)(rowBase + n16) * KD + 2 * half;
    // B fragment sources: centroid rows n16, 16+n16, 32+n16, 48+n16.
    const float* cr[4];
    float cn[4];
#pragma unroll
    for (int t = 0; t < 4; ++t) {
        cr[t] = cent + (size_t)(t * 16 + n16) * KD + 2 * half;
        cn[t] = cnorm[t * 16 + n16];   // ||c_{t*16+n16}||^2 (precomputed)
    }

    v8f acc[4] = {};                   // four 16x16 f32 accumulators
    float zp = 0.f;                    // partial ||z_row||^2 (this half's K)

#pragma unroll 4
    for (int kb = 0; kb < KD; kb += 4) {
        v2f av = *(const v2f*)(zrow + kb);
        zp += av.x * av.x + av.y * av.y;
#pragma unroll
        for (int t = 0; t < 4; ++t) {
            v2f bv = *(const v2f*)(cr[t] + kb);
            // D = A(16x4 f32) * B(4x16 f32) + C  -> v_wmma_f32_16x16x4_f32
            acc[t] = __builtin_amdgcn_wmma_f32_16x16x4_f32(
                false, av, false, bv, (short)0, acc[t], false, false);
        }
    }

    // Each half saw K%4 in {0,1} or {2,3}; combine halves for the z norm.
    float znf = zp + __shfl_xor(zp, 16, 32);          // ||z_{rowBase+n16}||^2

    // D layout: VGPR r, lane L -> row (r + 8*(L>>4)), col (L&15).
    float qv[4][8];
#pragma unroll
    for (int r = 0; r < 8; ++r) {
        // z-norm for THIS lane's row: fetch from the lane whose n16 == row idx.
        float znr = __shfl(znf, r + 8 * half, 32);
#pragma unroll
        for (int t = 0; t < 4; ++t) {
            float d  = acc[t][r];
            float s2 = fmaxf(znr + cn[t] - 2.0f * d, 0.0f);
            float sim = sqrtf(s2);
            qv[t][r] = 1.0f / (1.0f + sim);           // alpha = 1 Student-t
        }
        // Row sum across all 64 columns: 4 tiles + 16 lanes of this half.
        float rs = qv[0][r] + qv[1][r] + qv[2][r] + qv[3][r];
        rs += __shfl_xor(rs, 1, 32);
        rs += __shfl_xor(rs, 2, 32);
        rs += __shfl_xor(rs, 4, 32);
        rs += __shfl_xor(rs, 8, 32);                  // stays within the half
        float inv = 1.0f / rs;
#pragma unroll
        for (int t = 0; t < 4; ++t) {
            qv[t][r] *= inv;
            Qout[(size_t)(rowBase + r + 8 * half) * NC + t * 16 + n16] = qv[t][r];
        }
    }

    // Column partial sums over this wave's 16 rows (deterministic reduction).
#pragma unroll
    for (int t = 0; t < 4; ++t) {
        float cs = 0.f;
#pragma unroll
        for (int r = 0; r < 8; ++r) cs += qv[t][r];
        cs += __shfl_xor(cs, 16, 32);                 // combine the two halves
        if (lane < 16) csum[wave][t * 16 + n16] = cs;
    }
    __syncthreads();
    if (threadIdx.x < NC) {
        float s = 0.f;
#pragma unroll
        for (int w = 0; w < WAVES; ++w) s += csum[w][threadIdx.x];
        blockpart[(size_t)blockIdx.x * NC + threadIdx.x] = s;
    }
}

// ---------------------------------------------------------------------------
// Kernel 2: reduce per-block column partials -> colsum[64] (deterministic)
// ---------------------------------------------------------------------------
__global__ void __launch_bounds__(64)
clusternet_colsum_kernel(const float* __restrict__ blockpart,
                         float* __restrict__ colsum) {
    float s = 0.f;
#pragma unroll 8
    for (int b = 0; b < QBLK; ++b) s += blockpart[(size_t)b * NC + threadIdx.x];
    colsum[threadIdx.x] = s;
}

// ---------------------------------------------------------------------------
// Kernel 3: P = rownorm(Q^2 / colsum). One wave per row, 2 columns per lane.
// ---------------------------------------------------------------------------
__global__ void __launch_bounds__(256)
clusternet_p_kernel(const float* __restrict__ Q,
                    const float* __restrict__ colsum,
                    float* __restrict__ Pout) {
    const int lane = threadIdx.x & 31;
    const int row  = blockIdx.x * WAVES + (threadIdx.x >> 5);
    const size_t base = (size_t)row * NC;

    float q0 = Q[base + lane];
    float q1 = Q[base + lane + 32];
    float p0 = q0 * q0 / colsum[lane];
    float p1 = q1 * q1 / colsum[lane + 32];

    float s = p0 + p1;
    s += __shfl_xor(s, 1, 32);
    s += __shfl_xor(s, 2, 32);
    s += __shfl_xor(s, 4, 32);
    s += __shfl_xor(s, 8, 32);
    s += __shfl_xor(s, 16, 32);
    float inv = 1.0f / s;

    Pout[base + lane]      = p0 * inv;
    Pout[base + lane + 32] = p1 * inv;
}

// ---------------------------------------------------------------------------
extern "C" void kernel_launch(void* const* d_in, const int* in_sizes, int n_in,
                              void* d_out, int out_size, void* d_ws, size_t ws_size,
                              hipStream_t stream) {
    const float* z    = (const float*)d_in[0];   // 8192 x 256
    const float* cent = (const float*)d_in[1];   //   64 x 256

    float* Q = (float*)d_out;                    // first output  (8192 x 64)
    float* P = Q + (size_t)NR * NC;              // second output (8192 x 64)

    float* colsum    = (float*)d_ws;             // 64 floats
    float* blockpart = colsum + NC;              // QBLK * 64 floats (16 KB)
    float* cnorm     = blockpart + (size_t)QBLK * NC;  // 64 floats

    clusternet_cnorm_kernel<<<NC / WAVES, 256, 0, stream>>>(cent, cnorm);
    clusternet_q_kernel<<<QBLK, 256, 0, stream>>>(z, cent, cnorm, Q, blockpart);
    clusternet_colsum_kernel<<<1, 64, 0, stream>>>(blockpart, colsum);
    clusternet_p_kernel<<<NR / WAVES, 256, 0, stream>>>(Q, colsum, P);
}